// GCNNet_51943334478034
// MI455X (gfx1250) — compile-verified
//
#include <hip/hip_runtime.h>

typedef float v2f __attribute__((ext_vector_type(2)));
typedef float v8f __attribute__((ext_vector_type(8)));

#define N_NODES 100000
#define N_EDGES 1600000
#define IN_DIM  256
#define HID     128
#define OUT_DIM 40

// ---------------------------------------------------------------- degree pass
__global__ void k_init_deg(float* __restrict__ deg) {
    int i = blockIdx.x * blockDim.x + threadIdx.x;
    if (i < N_NODES) deg[i] = 1.0f;
}

__global__ void k_count_deg(const int* __restrict__ dst, float* __restrict__ deg) {
    int e = blockIdx.x * blockDim.x + threadIdx.x;
    if (e < N_EDGES) atomicAdd(&deg[dst[e]], 1.0f);
}

__global__ void k_rsqrt(float* __restrict__ deg) {
    int i = blockIdx.x * blockDim.x + threadIdx.x;
    if (i < N_NODES) deg[i] = rsqrtf(deg[i]);
}

// ---------------------------------------------------------------- GEMM1: h1 = x @ W1
// block = 256 threads (8 waves). Each block: 16-row M tile, wave w -> cols [16w,16w+16).
// A tile (16x256 f32) staged in LDS, row stride 260 -> conflict-free b64 reads.
#define XS_STRIDE 260
__launch_bounds__(256)
__global__ void k_gemm1(const float* __restrict__ x, const float* __restrict__ W,
                        float* __restrict__ h) {
    __shared__ float xs[16 * XS_STRIDE];
    const int tid  = threadIdx.x;
    const int lane = tid & 31;
    const int wave = tid >> 5;
    const int row0 = blockIdx.x * 16;

    // cooperative load of 16x256 A tile (16 floats per thread, float4 x4)
    {
        const int r = tid >> 4;           // 0..15
        const int c = (tid & 15) * 16;    // 0..240
        const float4* s = (const float4*)(x + (size_t)(row0 + r) * IN_DIM + c);
        float4* d       = (float4*)(xs + r * XS_STRIDE + c);
        d[0] = s[0]; d[1] = s[1]; d[2] = s[2]; d[3] = s[3];
    }
    __syncthreads();

    const int ln16 = lane & 15;           // M for A-frag, N for B-frag
    const int koff = (lane >> 4) * 2;     // K pair selector
    const int nb   = wave * 16;

    v8f c = {};
#pragma unroll 4
    for (int k0 = 0; k0 < IN_DIM; k0 += 4) {
        v2f a = *(const v2f*)(xs + ln16 * XS_STRIDE + k0 + koff);
        v2f b;
        b.x = W[(size_t)(k0 + koff)     * HID + nb + ln16];
        b.y = W[(size_t)(k0 + koff + 1) * HID + nb + ln16];
        c = __builtin_amdgcn_wmma_f32_16x16x4_f32(false, a, false, b,
                                                  (short)0, c, false, false);
    }
    // C/D layout: VGPR v -> row v + 8*(lane/16), col lane%16
    const int rbase = row0 + (lane >> 4) * 8;
#pragma unroll
    for (int v = 0; v < 8; ++v)
        h[(size_t)(rbase + v) * HID + nb + ln16] = c[v];
}

// ---------------------------------------------------------------- self-loop + bias init
__global__ void k_self1(const float* __restrict__ h1, const float* __restrict__ dinv,
                        const float* __restrict__ b1, float* __restrict__ agg) {
    int i = blockIdx.x * blockDim.x + threadIdx.x;
    if (i < N_NODES * HID) {
        int node = i >> 7;                 // /128
        float d = dinv[node];
        agg[i] = h1[i] * d * d + b1[i & 127];
    }
}

// ---------------------------------------------------------------- edge scatter, 128 dims
__launch_bounds__(256)
__global__ void k_scatter1(const int* __restrict__ src, const int* __restrict__ dst,
                           const float* __restrict__ dinv, const float* __restrict__ h1,
                           float* __restrict__ agg) {
    const int wave = threadIdx.x >> 5;
    const int lane = threadIdx.x & 31;
    const long e = (long)blockIdx.x * 8 + wave;
    if (e >= N_EDGES) return;
    const int s = src[e], d = dst[e];
    const float norm = dinv[s] * dinv[d];
    const float* hs = h1 + (size_t)s * HID;
    float* ad       = agg + (size_t)d * HID;
#pragma unroll
    for (int t = 0; t < 4; ++t) {
        int j = lane + 32 * t;
        atomicAdd(&ad[j], hs[j] * norm);
    }
}

// ---------------------------------------------------------------- GEMM2: h2 = relu(agg) @ W2
// block = 96 threads (3 waves); N=40 covered by 3 guarded 16-col tiles; ReLU fused in A load.
#define AS_STRIDE 132
__launch_bounds__(96)
__global__ void k_gemm2(const float* __restrict__ a_in, const float* __restrict__ W,
                        float* __restrict__ h) {
    __shared__ float as[16 * AS_STRIDE];
    const int tid  = threadIdx.x;
    const int lane = tid & 31;
    const int wave = tid >> 5;
    const int row0 = blockIdx.x * 16;

    for (int idx = tid; idx < 16 * HID; idx += 96) {
        int r = idx >> 7, cc = idx & 127;
        as[r * AS_STRIDE + cc] = fmaxf(a_in[(size_t)(row0 + r) * HID + cc], 0.0f);
    }
    __syncthreads();

    const int ln16 = lane & 15;
    const int koff = (lane >> 4) * 2;
    const int n    = wave * 16 + ln16;
    const bool nv  = (n < OUT_DIM);

    v8f c = {};
#pragma unroll 4
    for (int k0 = 0; k0 < HID; k0 += 4) {
        v2f a = *(const v2f*)(as + ln16 * AS_STRIDE + k0 + koff);
        v2f b;
        b.x = nv ? W[(size_t)(k0 + koff)     * OUT_DIM + n] : 0.0f;
        b.y = nv ? W[(size_t)(k0 + koff + 1) * OUT_DIM + n] : 0.0f;
        c = __builtin_amdgcn_wmma_f32_16x16x4_f32(false, a, false, b,
                                                  (short)0, c, false, false);
    }
    if (nv) {
        const int rbase = row0 + (lane >> 4) * 8;
#pragma unroll
        for (int v = 0; v < 8; ++v)
            h[(size_t)(rbase + v) * OUT_DIM + n] = c[v];
    }
}

__global__ void k_self2(const float* __restrict__ h2, const float* __restrict__ dinv,
                        const float* __restrict__ b2, float* __restrict__ out) {
    int i = blockIdx.x * blockDim.x + threadIdx.x;
    if (i < N_NODES * OUT_DIM) {
        int node = i / OUT_DIM;
        int j    = i - node * OUT_DIM;
        float d = dinv[node];
        out[i] = h2[i] * d * d + b2[j];
    }
}

// ---------------------------------------------------------------- edge scatter, 40 dims
__launch_bounds__(256)
__global__ void k_scatter2(const int* __restrict__ src, const int* __restrict__ dst,
                           const float* __restrict__ dinv, const float* __restrict__ h2,
                           float* __restrict__ out) {
    const int wave = threadIdx.x >> 5;
    const int lane = threadIdx.x & 31;
    const long e = (long)blockIdx.x * 8 + wave;
    if (e >= N_EDGES) return;
    const int s = src[e], d = dst[e];
    const float norm = dinv[s] * dinv[d];
    const float* hs = h2 + (size_t)s * OUT_DIM;
    float* od       = out + (size_t)d * OUT_DIM;
    if (lane < OUT_DIM)      atomicAdd(&od[lane],      hs[lane]      * norm);
    if (lane + 32 < OUT_DIM) atomicAdd(&od[lane + 32], hs[lane + 32] * norm);
}

// ---------------------------------------------------------------- launch
extern "C" void kernel_launch(void* const* d_in, const int* in_sizes, int n_in,
                              void* d_out, int out_size, void* d_ws, size_t ws_size,
                              hipStream_t stream) {
    const float* x  = (const float*)d_in[0];
    const int*   ei = (const int*)d_in[1];
    const float* W1 = (const float*)d_in[2];
    const float* b1 = (const float*)d_in[3];
    const float* W2 = (const float*)d_in[4];
    const float* b2 = (const float*)d_in[5];
    float* out = (float*)d_out;

    const int* src = ei;             // edge_index[0]
    const int* dst = ei + N_EDGES;   // edge_index[1]

    float* ws   = (float*)d_ws;
    float* dinv = ws;                                    // 100000 floats
    float* buf1 = ws + 100352;                           // h1, later reused for h2
    float* buf2 = buf1 + (size_t)N_NODES * HID;          // agg1 (relu input)

    k_init_deg <<<(N_NODES + 255) / 256, 256, 0, stream>>>(dinv);
    k_count_deg<<<(N_EDGES + 255) / 256, 256, 0, stream>>>(dst, dinv);
    k_rsqrt    <<<(N_NODES + 255) / 256, 256, 0, stream>>>(dinv);

    k_gemm1   <<<N_NODES / 16, 256, 0, stream>>>(x, W1, buf1);
    k_self1   <<<(N_NODES * HID + 255) / 256, 256, 0, stream>>>(buf1, dinv, b1, buf2);
    k_scatter1<<<N_EDGES / 8, 256, 0, stream>>>(src, dst, dinv, buf1, buf2);

    k_gemm2   <<<N_NODES / 16, 96, 0, stream>>>(buf2, W2, buf1);   // buf1 now = h2
    k_self2   <<<(N_NODES * OUT_DIM + 255) / 256, 256, 0, stream>>>(buf1, dinv, b2, out);
    k_scatter2<<<N_EDGES / 8, 256, 0, stream>>>(src, dst, dinv, buf1, out);
}